// QuanvolutionFilterQuantum_65481071405611
// MI455X (gfx1250) — compile-verified
//
#include <hip/hip_runtime.h>

typedef float v2f __attribute__((ext_vector_type(2)));
typedef float v8f __attribute__((ext_vector_type(8)));

#define N_PATCH 196

// One wave (32 lanes) processes 16 (batch,patch) states per tile iteration.
// D[e,n] = sum_d U[e,d] * amp[d,n] via 4x chained V_WMMA_F32_16X16X4_F32.
__global__ __launch_bounds__(256) void quanv_wmma_kernel(
    const float* __restrict__ x,   // [B, 784]
    const float* __restrict__ U,   // [16, 16] row-major U[e][d]
    float* __restrict__ out,       // [B, 784] = [B, 196*4]
    int nStates, int nTiles, int nWaves) {

  const int tid    = blockIdx.x * blockDim.x + threadIdx.x;
  const int lane   = threadIdx.x & 31;
  const int waveId = tid >> 5;
  const int row    = lane & 15;          // M for A, N (state column) for B/D
  const int hi     = lane >> 4;          // lane half: selects K/e sub-range

  // ---- A fragments: U[row, 4k + 2*hi + {0,1}] (f32 16x4 A layout), loaded once
  v2f a[4];
#pragma unroll
  for (int k = 0; k < 4; ++k) {
    const float* p = U + row * 16 + 4 * k + 2 * hi;
    a[k].x = p[0];
    a[k].y = p[1];
  }

  for (int tile = waveId; tile < nTiles; tile += nWaves) {
    int g  = tile * 16 + row;                    // global state index (b*196+p)
    int gc = (g < nStates) ? g : (nStates - 1);  // clamp tail (loads only)

    // decode (b, patch) -> pixel base
    unsigned ub = (unsigned)gc / N_PATCH;
    unsigned up = (unsigned)gc % N_PATCH;
    unsigned pr = up / 14u, pc = up % 14u;
    const float* px = x + (size_t)ub * 784u + pr * 56u + pc * 2u;

    // 2x2 patch pixels (8B-aligned float2 pairs)
    float2 t0 = *(const float2*)(px);        // x0, x1
    float2 t1 = *(const float2*)(px + 28);   // x2, x3

    float s0, c0, s1, c1, s2, c2, s3, c3;
    __sincosf(0.5f * t0.x, &s0, &c0);
    __sincosf(0.5f * t0.y, &s1, &c1);
    __sincosf(0.5f * t1.x, &s2, &c2);
    __sincosf(0.5f * t1.y, &s3, &c3);

    // amp[d], d = 4k + 2*hi + v :  q0[k>>1]*q1[k&1] * q2[hi] * q3[v]
    const float f2  = hi ? s2 : c2;
    const float pa0 = c0 * c1, pa1 = c0 * s1, pa2 = s0 * c1, pa3 = s0 * s1;
    const float g0 = pa0 * f2, g1 = pa1 * f2, g2 = pa2 * f2, g3 = pa3 * f2;

    v2f b0, b1, b2, b3;
    b0.x = g0 * c3; b0.y = g0 * s3;
    b1.x = g1 * c3; b1.y = g1 * s3;
    b2.x = g2 * c3; b2.y = g2 * s3;
    b3.x = g3 * c3; b3.y = g3 * s3;

    // ---- state = U * amp : 4 chained f32 WMMAs (K = 0..3,4..7,8..11,12..15)
    v8f acc = {};
    acc = __builtin_amdgcn_wmma_f32_16x16x4_f32(false, a[0], false, b0,
                                                (short)0, acc, false, false);
    acc = __builtin_amdgcn_wmma_f32_16x16x4_f32(false, a[1], false, b1,
                                                (short)0, acc, false, false);
    acc = __builtin_amdgcn_wmma_f32_16x16x4_f32(false, a[2], false, b2,
                                                (short)0, acc, false, false);
    acc = __builtin_amdgcn_wmma_f32_16x16x4_f32(false, a[3], false, b3,
                                                (short)0, acc, false, false);

    // ---- PauliZ expectations: lane holds state[e], e = r + 8*hi, r=0..7
    float sq0 = acc[0] * acc[0], sq1 = acc[1] * acc[1];
    float sq2 = acc[2] * acc[2], sq3 = acc[3] * acc[3];
    float sq4 = acc[4] * acc[4], sq5 = acc[5] * acc[5];
    float sq6 = acc[6] * acc[6], sq7 = acc[7] * acc[7];

    float lo4  = sq0 + sq1 + sq2 + sq3;
    float hi4  = sq4 + sq5 + sq6 + sq7;
    float sAll = lo4 + hi4;

    // w=0: sign = bit3(e) = hi (whole lane half)
    float m0 = hi ? -sAll : sAll;
    // w=1: sign = bit2(r)
    float m1 = lo4 - hi4;
    // w=2: sign = bit1(r)
    float m2 = (sq0 + sq1 + sq4 + sq5) - (sq2 + sq3 + sq6 + sq7);
    // w=3: sign = bit0(r)
    float m3 = (sq0 + sq2 + sq4 + sq6) - (sq1 + sq3 + sq5 + sq7);

    // combine the two e-halves (lane L <-> lane L+16)
    m0 += __shfl_xor(m0, 16, 32);
    m1 += __shfl_xor(m1, 16, 32);
    m2 += __shfl_xor(m2, 16, 32);
    m3 += __shfl_xor(m3, 16, 32);

    if (!hi && g < nStates) {
      float4 o = make_float4(m0, m1, m2, m3);
      *(float4*)(out + (size_t)g * 4u) = o;   // coalesced 256B per half-wave
    }
  }
}

extern "C" void kernel_launch(void* const* d_in, const int* in_sizes, int n_in,
                              void* d_out, int out_size, void* d_ws, size_t ws_size,
                              hipStream_t stream) {
  const float* x = (const float*)d_in[0];
  const float* U = (const float*)d_in[1];
  float* out     = (float*)d_out;

  const int batch   = in_sizes[0] / 784;
  const long long nStatesLL = (long long)batch * N_PATCH;
  const int nStates = (int)nStatesLL;
  const int nTiles  = (int)((nStatesLL + 15) / 16);

  const int threads = 256;                 // 8 waves per block
  const int wpb     = threads / 32;
  int blocks        = 4096;                // grid-stride; ~12 tiles/wave at B=32768
  const int maxBlocks = (nTiles + wpb - 1) / wpb;
  if (blocks > maxBlocks) blocks = maxBlocks;
  const int nWaves = blocks * wpb;

  quanv_wmma_kernel<<<blocks, threads, 0, stream>>>(x, U, out,
                                                    nStates, nTiles, nWaves);
}